// Model_41274635714637
// MI455X (gfx1250) — compile-verified
//
#include <hip/hip_runtime.h>
#include <math.h>

typedef float v2f __attribute__((ext_vector_type(2)));
typedef float v8f __attribute__((ext_vector_type(8)));

#define B_   16
#define N_   4096
#define C_   5
#define D_   64
#define S_   1024
#define K_   32
#define CD_  69      // D_ + C_
#define OC_  133     // CD_ + D_
#define EPSF 1e-5f

// ---------------------------------------------------------------------------
// Kernel 1: farthest point sampling. One 1024-thread block per batch.
// Each thread owns 4 points (coords + running min-distance live in registers).
// Matches lax.scan semantics: fps_idx[s] is the farthest BEFORE the update.
// ---------------------------------------------------------------------------
__global__ __launch_bounds__(1024) void fps_kernel(const float* __restrict__ xyz,
                                                   const int* __restrict__ finit,
                                                   int* __restrict__ fps_idx)
{
    const int b    = blockIdx.x;
    const int tid  = threadIdx.x;
    const int lane = tid & 31;

    __shared__ float cen[8];
    __shared__ float red_v[32];
    __shared__ int   red_i[32];
    __shared__ int   far_s;

    float px[4][5];
    float dmin[4];
#pragma unroll
    for (int r = 0; r < 4; ++r) {
        const int p = r * 1024 + tid;
        const float* q = xyz + ((size_t)b * N_ + p) * C_;
#pragma unroll
        for (int c = 0; c < 5; ++c) px[r][c] = q[c];
        dmin[r] = 1.0e10f;
    }
    if (tid == 0) far_s = finit[b];
    __syncthreads();

    for (int s = 0; s < S_; ++s) {
        const int far = far_s;
        if (tid == 0) fps_idx[b * S_ + s] = far;
        if (tid < 5)  cen[tid] = xyz[((size_t)b * N_ + far) * C_ + tid];
        __syncthreads();
        const float c0 = cen[0], c1 = cen[1], c2 = cen[2], c3 = cen[3], c4 = cen[4];

        float bv = -1.0f; int bi = 0;
#pragma unroll
        for (int r = 0; r < 4; ++r) {
            const float d0 = px[r][0] - c0, d1 = px[r][1] - c1, d2 = px[r][2] - c2,
                        d3 = px[r][2+1] - c3, d4 = px[r][4] - c4;
            const float dd = d0*d0 + d1*d1 + d2*d2 + d3*d3 + d4*d4;
            dmin[r] = fminf(dmin[r], dd);
            const int p = r * 1024 + tid;
            if (dmin[r] > bv) { bv = dmin[r]; bi = p; }
        }
        // wave32 argmax (tie -> smaller index, matching jnp.argmax first-hit)
#pragma unroll
        for (int off = 16; off > 0; off >>= 1) {
            const float ov = __shfl_xor(bv, off, 32);
            const int   oi = __shfl_xor(bi, off, 32);
            if (ov > bv || (ov == bv && oi < bi)) { bv = ov; bi = oi; }
        }
        if (lane == 0) { red_v[tid >> 5] = bv; red_i[tid >> 5] = bi; }
        __syncthreads();
        if (tid < 32) {
            bv = red_v[tid]; bi = red_i[tid];
#pragma unroll
            for (int off = 16; off > 0; off >>= 1) {
                const float ov = __shfl_xor(bv, off, 32);
                const int   oi = __shfl_xor(bi, off, 32);
                if (ov > bv || (ov == bv && oi < bi)) { bv = ov; bi = oi; }
            }
            if (tid == 0) far_s = bi;
        }
        __syncthreads();
    }
}

// ---------------------------------------------------------------------------
// Kernel 2: gather new_xyz into the head of d_out.
// ---------------------------------------------------------------------------
__global__ void newxyz_kernel(const float* __restrict__ xyz,
                              const int* __restrict__ fps_idx,
                              float* __restrict__ out)
{
    const int i = blockIdx.x * blockDim.x + threadIdx.x;
    if (i < B_ * S_ * C_) {
        const int c = i % C_;
        const int s = (i / C_) % S_;
        const int b = i / (C_ * S_);
        out[i] = xyz[((size_t)b * N_ + fps_idx[b * S_ + s]) * C_ + c];
    }
}

// ---------------------------------------------------------------------------
// Kernel 3: KNN. Block = 512 threads (16 waves) handles 16 queries of one
// batch. Phase 1: distance tile via V_WMMA_F32_16X16X4_F32, with |q|^2+|p|^2
// folded into C and -2 folded into A. The full 16x4096 f32 distance tile is
// held in LDS (256 KB — CDNA5's 320 KB/WGP makes this possible and avoids
// ~256 MB of HBM spill traffic). Phase 2: one wave per query does 32 rounds
// of wave-min selection out of LDS (ascending distance == top_k(-sq) order).
// ---------------------------------------------------------------------------
__global__ __launch_bounds__(512) void knn_kernel(const float* __restrict__ xyz,
                                                  const int* __restrict__ fps_idx,
                                                  int* __restrict__ knn_idx)
{
    extern __shared__ float smem[];
    float* qcoord = smem;         // 16 queries * 8 padded coords
    float* qn2    = smem + 128;   // 16 query squared norms
    float* dist   = smem + 160;   // 16 * 4096 distance tile

    const int b    = blockIdx.x;
    const int s0   = blockIdx.y * 16;
    const int tid  = threadIdx.x;
    const int wave = tid >> 5;
    const int lane = tid & 31;

    if (tid < 128) {
        const int q = tid >> 3, c = tid & 7;
        const int qi = fps_idx[b * S_ + s0 + q];
        qcoord[q * 8 + c] = (c < 5) ? xyz[((size_t)b * N_ + qi) * C_ + c] : 0.0f;
    }
    __syncthreads();
    if (tid < 16) {
        float s = 0.f;
#pragma unroll
        for (int c = 0; c < 5; ++c) { const float v = qcoord[tid * 8 + c]; s += v * v; }
        qn2[tid] = s;
    }
    __syncthreads();

    // A-fragment (16x4 f32): lanes 0-15 hold K=0,1; lanes 16-31 hold K=2,3.
    const int  m  = lane & 15;
    const bool hi = lane >= 16;
    const int  mo = hi ? 8 : 0;   // C/D: M = mo + r, N = lane & 15
    const float q0 = qcoord[m*8+0], q1 = qcoord[m*8+1], q2 = qcoord[m*8+2],
                q3 = qcoord[m*8+3], q4 = qcoord[m*8+4];
    v2f a0, a1;
    a0.x = -2.f * (hi ? q2 : q0);  a0.y = -2.f * (hi ? q3 : q1);
    a1.x = hi ? 0.f : (-2.f * q4); a1.y = 0.f;   // K=4..7: only coord 4 non-zero

    for (int tile = wave; tile < N_ / 16; tile += 16) {   // uniform: EXEC all-1s
        const int n0 = tile * 16;
        const float* p = xyz + ((size_t)b * N_ + n0 + m) * C_;
        const float c0 = p[0], c1 = p[1], c2 = p[2], c3 = p[3], c4 = p[4];
        const float pn2 = c0*c0 + c1*c1 + c2*c2 + c3*c3 + c4*c4;
        v2f b0, b1;
        b0.x = hi ? c2 : c0;  b0.y = hi ? c3 : c1;
        b1.x = hi ? 0.f : c4; b1.y = 0.f;
        v8f acc;
#pragma unroll
        for (int r = 0; r < 8; ++r) acc[r] = qn2[mo + r] + pn2;
        acc = __builtin_amdgcn_wmma_f32_16x16x4_f32(false, a0, false, b0,
                                                    (short)0, acc, false, false);
        acc = __builtin_amdgcn_wmma_f32_16x16x4_f32(false, a1, false, b1,
                                                    (short)0, acc, false, false);
#pragma unroll
        for (int r = 0; r < 8; ++r) dist[(mo + r) * N_ + n0 + m] = acc[r];
    }
    __syncthreads();

    // Phase 2: iterative top-32 (ascending distance, tie -> smaller index).
    float* row  = dist + wave * N_;
    int*   kout = knn_idx + ((size_t)b * S_ + s0 + wave) * K_;
    for (int k = 0; k < K_; ++k) {
        float bv = 3.0e38f; int bi = N_;
        for (int j = lane; j < N_; j += 32) {
            const float v = row[j];
            if (v < bv) { bv = v; bi = j; }
        }
#pragma unroll
        for (int off = 16; off > 0; off >>= 1) {
            const float ov = __shfl_xor(bv, off, 32);
            const int   oi = __shfl_xor(bi, off, 32);
            if (ov < bv || (ov == bv && oi < bi)) { bv = ov; bi = oi; }
        }
        if (lane == 0) { kout[k] = bi; row[bi] = 3.0e38f; }
        __syncthreads();   // uniform across all 16 waves
    }
}

// ---------------------------------------------------------------------------
// Kernel 4: per-group channel means + per-batch centered-sum-of-squares.
// One 128-thread block per (b,s) group; channels map to threads.
// ---------------------------------------------------------------------------
__global__ __launch_bounds__(128) void stats_kernel(const float* __restrict__ xyz,
                                                    const float* __restrict__ points,
                                                    const int* __restrict__ knn_idx,
                                                    float* __restrict__ meanw,
                                                    float* __restrict__ varacc)
{
    const int g   = blockIdx.x;        // b*S + s
    const int b   = g >> 10;
    const int tid = threadIdx.x;
    __shared__ int   idx[32];
    __shared__ float red[128];
    if (tid < 32) idx[tid] = knn_idx[(size_t)g * K_ + tid];
    __syncthreads();

    float sq = 0.f;
    if (tid < CD_) {
        const int c = tid;
        float sum = 0.f;
#pragma unroll 4
        for (int k = 0; k < K_; ++k) {
            const int p = idx[k];
            const float v = (c < D_) ? points[((size_t)b * N_ + p) * D_ + c]
                                     : xyz[((size_t)b * N_ + p) * C_ + (c - D_)];
            sum += v;
        }
        const float mean = sum * (1.0f / K_);
        meanw[(size_t)g * CD_ + c] = mean;
#pragma unroll 4
        for (int k = 0; k < K_; ++k) {
            const int p = idx[k];
            const float v = (c < D_) ? points[((size_t)b * N_ + p) * D_ + c]
                                     : xyz[((size_t)b * N_ + p) * C_ + (c - D_)];
            const float d = v - mean;
            sq += d * d;
        }
    }
    red[tid] = sq;
    __syncthreads();
    for (int off = 64; off > 0; off >>= 1) {
        if (tid < off) red[tid] += red[tid + off];
        __syncthreads();
    }
    if (tid == 0) atomicAdd(varacc + b, red[0]);
}

__global__ void zero_kernel(float* __restrict__ varacc)
{
    if (threadIdx.x < B_) varacc[threadIdx.x] = 0.f;
}

__global__ void std_kernel(const float* __restrict__ varacc, float* __restrict__ stdv)
{
    const int b = threadIdx.x;
    if (b < B_) {
        const float n1 = (float)((size_t)S_ * K_ * CD_ - 1);   // ddof=1
        stdv[b] = sqrtf(varacc[b] / n1);
    }
}

// ---------------------------------------------------------------------------
// Kernel 5: fused gather + normalize + affine + broadcast-concat output.
// Memory-bound (~560 MB total traffic -> the real cost of this op).
// ---------------------------------------------------------------------------
__global__ __launch_bounds__(256) void out_kernel(const float* __restrict__ xyz,
                                                  const float* __restrict__ points,
                                                  const float* __restrict__ alpha,
                                                  const float* __restrict__ beta,
                                                  const int* __restrict__ fps_idx,
                                                  const int* __restrict__ knn_idx,
                                                  const float* __restrict__ meanw,
                                                  const float* __restrict__ stdv,
                                                  float* __restrict__ out)
{
    const size_t total = (size_t)B_ * S_ * K_ * OC_;
    for (size_t i = (size_t)blockIdx.x * blockDim.x + threadIdx.x; i < total;
         i += (size_t)gridDim.x * blockDim.x) {
        const int c = (int)(i % OC_);
        size_t t = i / OC_;
        const int k = (int)(t % K_); t /= K_;
        const int s = (int)(t % S_); t /= S_;
        const int b = (int)t;
        float val;
        if (c < CD_) {
            const int p = knn_idx[((size_t)b * S_ + s) * K_ + k];
            const float g = (c < D_) ? points[((size_t)b * N_ + p) * D_ + c]
                                     : xyz[((size_t)b * N_ + p) * C_ + (c - D_)];
            const float mn = meanw[((size_t)b * S_ + s) * CD_ + c];
            val = alpha[c] * ((g - mn) / (stdv[b] + EPSF)) + beta[c];
        } else {
            const int qi = fps_idx[(size_t)b * S_ + s];
            val = points[((size_t)b * N_ + qi) * D_ + (c - CD_)];
        }
        out[i] = val;
    }
}

// ---------------------------------------------------------------------------
extern "C" void kernel_launch(void* const* d_in, const int* in_sizes, int n_in,
                              void* d_out, int out_size, void* d_ws, size_t ws_size,
                              hipStream_t stream)
{
    const float* xyz    = (const float*)d_in[0];
    const float* points = (const float*)d_in[1];
    const float* alpha  = (const float*)d_in[2];
    const float* beta   = (const float*)d_in[3];
    const int*   finit  = (const int*)d_in[4];
    float* out = (float*)d_out;

    char* ws = (char*)d_ws;
    int*   fps    = (int*)ws;    ws += (((size_t)B_ * S_ * 4) + 255) & ~(size_t)255;
    int*   knn    = (int*)ws;    ws += (((size_t)B_ * S_ * K_ * 4) + 255) & ~(size_t)255;
    float* meanw  = (float*)ws;  ws += (((size_t)B_ * S_ * CD_ * 4) + 255) & ~(size_t)255;
    float* varacc = (float*)ws;  ws += 256;
    float* stdv   = (float*)ws;

    fps_kernel<<<B_, 1024, 0, stream>>>(xyz, finit, fps);
    newxyz_kernel<<<(B_ * S_ * C_ + 255) / 256, 256, 0, stream>>>(xyz, fps, out);

    dim3 kgrid(B_, S_ / 16);
    const size_t lds_bytes = (size_t)(160 + 16 * N_) * sizeof(float);  // ~257 KB
    knn_kernel<<<kgrid, 512, lds_bytes, stream>>>(xyz, fps, knn);

    zero_kernel<<<1, 32, 0, stream>>>(varacc);
    stats_kernel<<<B_ * S_, 128, 0, stream>>>(xyz, points, knn, meanw, varacc);
    std_kernel<<<1, 32, 0, stream>>>(varacc, stdv);

    float* out2 = out + (size_t)B_ * S_ * C_;
    const size_t total = (size_t)B_ * S_ * K_ * OC_;
    const int blocks = (int)((total + 255) / 256);
    out_kernel<<<blocks, 256, 0, stream>>>(xyz, points, alpha, beta, fps, knn,
                                           meanw, stdv, out2);
}